// mIoUEstimator_22247930593534
// MI455X (gfx1250) — compile-verified
//
#include <hip/hip_runtime.h>

// ---- problem constants (fixed by the reference) ----
#define N_CLASSES 91
#define NBINS     96          // padded to multiple of 4 (WMMA K-step) / nice LDS shape
#define NBATCH    32
#define NPB       (1024*1024) // pixels per batch image

// ---- histogram kernel tiling ----
#define THREADS   256
#define WAVES     8           // 256 / wave32
#define VEC       4           // 128-bit loads
#define ITERS     32
#define PIX_PER_BLOCK (THREADS*VEC*ITERS)       // 32768
#define BLOCKS_PER_BATCH (NPB / PIX_PER_BLOCK)  // 32
#define HIST_WORDS (3*NBATCH*NBINS)             // 9216 u32 in workspace

typedef __attribute__((ext_vector_type(2))) float v2f;
typedef __attribute__((ext_vector_type(8))) float v8f;
typedef __attribute__((ext_vector_type(4))) int   v4i;   // native vector: OK for nontemporal builtin

// ---------------------------------------------------------------------------
// Kernel 1: zero the global histogram workspace (harness poisons d_ws).
// ---------------------------------------------------------------------------
__global__ void zero_ws_kernel(unsigned* __restrict__ g) {
    int i = blockIdx.x * blockDim.x + threadIdx.x;
    if (i < HIST_WORDS) g[i] = 0u;
}

// ---------------------------------------------------------------------------
// Kernel 2: streaming histogram. Each block owns a 32K-pixel chunk of ONE
// batch image (batch index is block-uniform -> scalar). Per-wave privatized
// LDS histograms (ds_add_u32), then block reduction -> global_atomic_add_u32.
// ghist layout: [3][NBATCH][NBINS]  (0 = cnt_x, 1 = cnt_t, 2 = inter)
// ---------------------------------------------------------------------------
__global__ __launch_bounds__(THREADS) void
hist_kernel(const int* __restrict__ xs, const int* __restrict__ ts,
            unsigned* __restrict__ ghist) {
    __shared__ unsigned lh[WAVES][3][NBINS];

    const int tid  = threadIdx.x;
    const int wave = tid >> 5;           // wave32

    for (int i = tid; i < WAVES*3*NBINS; i += THREADS)
        ((unsigned*)lh)[i] = 0u;
    __syncthreads();

    const int batch = blockIdx.x / BLOCKS_PER_BATCH;
    const int chunk = blockIdx.x % BLOCKS_PER_BATCH;
    const int base4 = (batch * NPB + chunk * PIX_PER_BLOCK) >> 2; // v4i index
    const v4i* __restrict__ x4 = ((const v4i*)xs) + base4;
    const v4i* __restrict__ t4 = ((const v4i*)ts) + base4;

    unsigned (* __restrict__ h)[NBINS] = lh[wave];

#pragma unroll 4
    for (int it = 0; it < ITERS; ++it) {
        const int idx = it * THREADS + tid;
        // streamed exactly once: non-temporal 128-bit loads
        v4i a = __builtin_nontemporal_load(&x4[idx]);
        v4i b = __builtin_nontemporal_load(&t4[idx]);

#define PROC(ax, bx)                                              \
        do {                                                      \
            unsigned ca = (unsigned)(ax), cb = (unsigned)(bx);    \
            atomicAdd(&h[0][ca], 1u);                             \
            atomicAdd(&h[1][cb], 1u);                             \
            if (ca == cb) atomicAdd(&h[2][ca], 1u);               \
        } while (0)

        PROC(a.x, b.x);
        PROC(a.y, b.y);
        PROC(a.z, b.z);
        PROC(a.w, b.w);
#undef PROC
    }
    __syncthreads();

    // reduce the 8 wave copies, push to global with one atomic per bin
    for (int j = tid; j < 3*NBINS; j += THREADS) {
        const int ht  = j / NBINS;
        const int bin = j % NBINS;
        unsigned s = 0;
#pragma unroll
        for (int w = 0; w < WAVES; ++w) s += lh[w][ht][bin];
        if (s) atomicAdd(&ghist[(ht * NBATCH + batch) * NBINS + bin], s);
    }
}

// ---------------------------------------------------------------------------
// Kernel 3: finalize. 2 waves, each owns 16 batches. Build the 16x96 IoU
// matrix 4 K-columns at a time and row-reduce with V_WMMA_F32_16X16X4_F32
// against an all-ones B (D[m][n] = sum_k A[m][k]).
// A layout (32-bit 16x4, wave32): lanes 0-15 hold M=lane, VGPR0=K0, VGPR1=K1;
// lanes 16-31 hold M=lane-16, VGPR0=K2, VGPR1=K3.
// D layout: lane0 acc[r] = D[r][0] (rows 0..7), lane16 acc[r] = D[r+8][0].
// ---------------------------------------------------------------------------
__global__ __launch_bounds__(64) void
iou_kernel(const unsigned* __restrict__ ghist,
           const float* __restrict__ smooth,
           float* __restrict__ out) {
    const float s    = smooth[0];
    const int   lane = threadIdx.x & 31;
    const int   wave = threadIdx.x >> 5;
    const int   m    = lane & 15;
    const int   b    = wave * 16 + m;        // batch row this lane feeds
    const int   koff = (lane >> 4) ? 2 : 0;  // lanes 16-31 carry K=2,3

    const unsigned* __restrict__ hx = ghist + (0 * NBATCH + b) * NBINS;
    const unsigned* __restrict__ ht = ghist + (1 * NBATCH + b) * NBINS;
    const unsigned* __restrict__ hi = ghist + (2 * NBATCH + b) * NBINS;

    v8f acc = {0.f, 0.f, 0.f, 0.f, 0.f, 0.f, 0.f, 0.f};
    v2f ones; ones.x = 1.0f; ones.y = 1.0f;  // every element of B is 1.0

    for (int k0 = 0; k0 < NBINS; k0 += 4) {
        v2f a;
#pragma unroll
        for (int j = 0; j < 2; ++j) {
            const int c = k0 + koff + j;
            const float xi = (float)hx[c];
            const float ti = (float)ht[c];
            const float ii = (float)hi[c];
            const float iou = (ii + s) / (xi + ti - ii + s);
            const float v = (c < N_CLASSES) ? iou : 0.0f;  // padded classes -> 0
            if (j == 0) a.x = v; else a.y = v;
        }
        acc = __builtin_amdgcn_wmma_f32_16x16x4_f32(
                  false, a, false, ones, (short)0, acc, false, false);
    }

    if (lane == 0) {
#pragma unroll
        for (int r = 0; r < 8; ++r)
            out[wave * 16 + r] = acc[r] / (float)N_CLASSES;
    } else if (lane == 16) {
#pragma unroll
        for (int r = 0; r < 8; ++r)
            out[wave * 16 + 8 + r] = acc[r] / (float)N_CLASSES;
    }
}

// ---------------------------------------------------------------------------
extern "C" void kernel_launch(void* const* d_in, const int* in_sizes, int n_in,
                              void* d_out, int out_size, void* d_ws, size_t ws_size,
                              hipStream_t stream) {
    const int*   x      = (const int*)d_in[0];
    const int*   t      = (const int*)d_in[1];
    const float* smooth = (const float*)d_in[2];
    unsigned*    ghist  = (unsigned*)d_ws;   // needs 3*32*96*4 = 36 KB
    float*       out    = (float*)d_out;

    zero_ws_kernel<<<(HIST_WORDS + 255) / 256, 256, 0, stream>>>(ghist);
    hist_kernel<<<NBATCH * BLOCKS_PER_BATCH, THREADS, 0, stream>>>(x, t, ghist);
    iou_kernel<<<1, 64, 0, stream>>>(ghist, smooth, out);
}